// CRF0_1649267441995
// MI455X (gfx1250) — compile-verified
//
#include <hip/hip_runtime.h>
#include <stddef.h>
#include <stdint.h>

#define SENT_LEN 2048
#define CLASS_NUM 4096

typedef float v2f __attribute__((ext_vector_type(2)));
typedef float v8f __attribute__((ext_vector_type(8)));
typedef int a_v4i __attribute__((vector_size(16)));   // matches builtin's pointee type

#ifndef __has_builtin
#define __has_builtin(x) 0
#endif

// ---------------- prep kernels (workspace path) ----------------

// expT = exp(T); expTt = transpose(expT)  (both C x C, row-major)
__global__ void k_prep_T(const float* __restrict__ T,
                         float* __restrict__ expT,
                         float* __restrict__ expTt) {
    size_t idx = (size_t)blockIdx.x * blockDim.x + threadIdx.x;
    size_t r = idx >> 12;            // / CLASS_NUM
    size_t c = idx & (CLASS_NUM - 1);
    float v = expf(T[idx]);
    expT[idx] = v;
    expTt[c * (size_t)CLASS_NUM + r] = v;
}

// exps = exp(scores); also emit alpha[0] and beta[L-1]
__global__ void k_prep_S(const float* __restrict__ scores,
                         float* __restrict__ exps,
                         float* __restrict__ alpha,
                         float* __restrict__ beta) {
    size_t idx = (size_t)blockIdx.x * blockDim.x + threadIdx.x;
    float v = expf(scores[idx]);
    exps[idx] = v;
    size_t row = idx >> 12;
    if (row == 0) alpha[idx] = v;                       // alpha row 0
    if (row == SENT_LEN - 1) beta[idx] = v;             // beta row L-1 (idx == (L-1)*C + c)
}

// ---------------- main step kernel: out[x] = e[x] * sum_y prev[y]*M[y][x] ----------------
// gridDim = (C/64, 2): blockIdx.y selects direction (0=fwd uses expT, 1=bwd uses expT^T).
// 256 threads = 8 waves. Waves (w&3) pick one of 4 x-tiles of 16; (w>>2) picks K half.
// GEMV mapped onto V_WMMA_F32_16X16X4_F32 with broadcast-A:
//   A[m][k] = prev[y+k]  (same for every m)  -> every row of D = 16-output partial sums.
__global__ __launch_bounds__(256)
void k_step(const float* __restrict__ Mf, const float* __restrict__ Mb,
            const float* __restrict__ pF, const float* __restrict__ pB,
            const float* __restrict__ eF, const float* __restrict__ eB,
            float* __restrict__ oF, float* __restrict__ oB) {
    __shared__ float sPrev[CLASS_NUM];   // 16 KB: prev vector, broadcast-reused by all waves
    __shared__ float sRed[128];          // per-wave 16-output partials for K-split reduce

    const int dir = blockIdx.y;
    const float* M    = dir ? Mb : Mf;
    const float* prev = dir ? pB : pF;
    const float* ev   = dir ? eB : eF;
    float*       outp = dir ? oB : oF;

    // Stage prev into LDS. Prefer the CDNA5 async-to-LDS engine (no VGPR round-trip,
    // tracked by ASYNCcnt); fall back to plain coalesced copies if unavailable.
#if __has_builtin(__builtin_amdgcn_global_load_async_to_lds_b128)
    {
        a_v4i* g = (a_v4i*)prev;     // generic pointers; codegen inserts AS casts
        a_v4i* l = (a_v4i*)sPrev;
        #pragma unroll
        for (int k = 0; k < 4; ++k) {
            int e = threadIdx.x + k * 256;            // 1024 x b128 = 16 KB
            __builtin_amdgcn_global_load_async_to_lds_b128(g + e, l + e, 0, 0);
        }
#if __has_builtin(__builtin_amdgcn_s_wait_asynccnt)
        __builtin_amdgcn_s_wait_asynccnt(0);
#else
        asm volatile("s_wait_asynccnt 0x0" ::: "memory");
#endif
    }
#else
    {
        const float4* p4 = (const float4*)prev;
        float4* s4 = (float4*)sPrev;
        for (int i = threadIdx.x; i < CLASS_NUM / 4; i += 256) s4[i] = p4[i];
    }
#endif
    __syncthreads();

    const int tid  = threadIdx.x;
    const int wave = tid >> 5;
    const int lane = tid & 31;
    const int half = lane >> 4;      // lanes 16-31 hold K+2 per 16x16x4 operand layout
    const int ln   = lane & 15;
    const int x0   = blockIdx.x * 64 + (wave & 3) * 16;  // global x base of this wave's tile
    const int y0   = (wave >> 2) * (CLASS_NUM / 2);      // 2-way K split

    // B operand pointer: M[y0 + half*2][x0 + ln]; second element is next row (+C).
    const float* Bp = M + (size_t)(y0 + half * 2) * CLASS_NUM + x0 + ln;
    const float* sp = sPrev + y0 + half * 2;

    float partial;
#if __has_builtin(__builtin_amdgcn_wmma_f32_16x16x4_f32)
    v8f acc = {0.f, 0.f, 0.f, 0.f, 0.f, 0.f, 0.f, 0.f};
    for (int kk = 0; kk < CLASS_NUM / 2; kk += 16) {
        __builtin_prefetch(Bp + (size_t)32 * CLASS_NUM, 0, 1);  // global_prefetch_b8, ~2 chunks ahead
        #pragma unroll
        for (int k = 0; k < 16; k += 4) {
            v2f a, b;
            a.x = sp[0];            // A: prev[y + klo],   klo = half*2
            a.y = sp[1];            //    prev[y + klo+1]
            b.x = Bp[0];            // B: M[y+klo  ][x0+ln]
            b.y = Bp[CLASS_NUM];    //    M[y+klo+1][x0+ln]
            acc = __builtin_amdgcn_wmma_f32_16x16x4_f32(
                false, a, false, b, (short)0, acc, false, false);
            sp += 4;
            Bp += (size_t)4 * CLASS_NUM;
        }
    }
    partial = acc[0];  // D row M=0 (all rows identical): lanes 0-15 hold the 16 outputs
#else
    // VALU fallback (also what the host-side parse sees)
    partial = 0.f;
    const float* col = M + x0 + ln;
    for (int k = 0; k < CLASS_NUM / 2; ++k)
        partial += sPrev[y0 + k] * col[(size_t)(y0 + k) * CLASS_NUM];
#endif

    if (half == 0) sRed[wave * 16 + ln] = partial;
    __syncthreads();
    if (tid < 64) {  // 4 tiles x 16 outputs; deterministic 2-way K reduce
        int i = tid >> 4, l = tid & 15;
        float s = sRed[i * 16 + l] + sRed[(i + 4) * 16 + l];
        int x = blockIdx.x * 64 + i * 16 + l;
        outp[x] = ev[x] * s;
    }
}

// ---------------- fallback path (workspace too small): exp on the fly ----------------

__global__ void k_init_fb(const float* __restrict__ scores,
                          float* __restrict__ alpha, float* __restrict__ beta) {
    int idx = blockIdx.x * blockDim.x + threadIdx.x;  // 0 .. 2C
    if (idx < CLASS_NUM) {
        alpha[idx] = expf(scores[idx]);
    } else {
        int c = idx - CLASS_NUM;
        size_t off = (size_t)(SENT_LEN - 1) * CLASS_NUM + c;
        beta[off] = expf(scores[off]);
    }
}

__global__ __launch_bounds__(256)
void k_fwd_fb(const float* __restrict__ T, const float* __restrict__ erow,
              const float* __restrict__ prev, float* __restrict__ out) {
    __shared__ float sPrev[CLASS_NUM];
    for (int i = threadIdx.x; i < CLASS_NUM; i += 256) sPrev[i] = prev[i];
    __syncthreads();
    int x = blockIdx.x * 256 + threadIdx.x;
    float acc = 0.f;
    for (int y = 0; y < CLASS_NUM; ++y)
        acc += sPrev[y] * expf(T[(size_t)y * CLASS_NUM + x]);
    out[x] = expf(erow[x]) * acc;
}

__global__ __launch_bounds__(256)
void k_bwd_fb(const float* __restrict__ T, const float* __restrict__ erow,
              const float* __restrict__ nxt, float* __restrict__ out) {
    __shared__ float sN[CLASS_NUM];
    for (int i = threadIdx.x; i < CLASS_NUM; i += 256) sN[i] = nxt[i];
    __syncthreads();
    int wave = threadIdx.x >> 5, lane = threadIdx.x & 31;
    int nw = (gridDim.x * 256) >> 5;
    for (int x = blockIdx.x * 8 + wave; x < CLASS_NUM; x += nw) {
        const float* row = T + (size_t)x * CLASS_NUM;
        float acc = 0.f;
        for (int y = lane; y < CLASS_NUM; y += 32) acc += expf(row[y]) * sN[y];
        for (int o = 16; o; o >>= 1) acc += __shfl_down(acc, o, 32);
        if (lane == 0) out[x] = expf(erow[x]) * acc;
    }
}

// ---------------- host ----------------

extern "C" void kernel_launch(void* const* d_in, const int* in_sizes, int n_in,
                              void* d_out, int out_size, void* d_ws, size_t ws_size,
                              hipStream_t stream) {
    const float* scores = (const float*)d_in[0];
    const float* T      = (const float*)d_in[1];
    float* out   = (float*)d_out;
    const size_t CC = (size_t)CLASS_NUM * CLASS_NUM;
    const size_t LC = (size_t)SENT_LEN * CLASS_NUM;
    float* alpha = out;
    float* beta  = out + LC;

    size_t need = (2 * CC + LC) * sizeof(float);  // expT + expT^T + exps = 160 MB (fits L2)
    if (d_ws && ws_size >= need) {
        float* expT  = (float*)d_ws;
        float* expTt = expT + CC;
        float* exps  = expTt + CC;
        k_prep_T<<<(unsigned)(CC / 256), 256, 0, stream>>>(T, expT, expTt);
        k_prep_S<<<(unsigned)(LC / 256), 256, 0, stream>>>(scores, exps, alpha, beta);
        for (int t = 1; t < SENT_LEN; ++t) {
            k_step<<<dim3(CLASS_NUM / 64, 2), 256, 0, stream>>>(
                expT, expTt,
                alpha + (size_t)(t - 1) * CLASS_NUM,          // fwd prev
                beta  + (size_t)(SENT_LEN - t) * CLASS_NUM,   // bwd prev (row j+1)
                exps  + (size_t)t * CLASS_NUM,                // fwd e
                exps  + (size_t)(SENT_LEN - 1 - t) * CLASS_NUM,
                alpha + (size_t)t * CLASS_NUM,                // fwd out
                beta  + (size_t)(SENT_LEN - 1 - t) * CLASS_NUM);
        }
    } else {
        k_init_fb<<<(2 * CLASS_NUM) / 256, 256, 0, stream>>>(scores, alpha, beta);
        for (int t = 1; t < SENT_LEN; ++t) {
            int j = SENT_LEN - 1 - t;
            k_fwd_fb<<<CLASS_NUM / 256, 256, 0, stream>>>(
                T, scores + (size_t)t * CLASS_NUM,
                alpha + (size_t)(t - 1) * CLASS_NUM,
                alpha + (size_t)t * CLASS_NUM);
            k_bwd_fb<<<64, 256, 0, stream>>>(
                T, scores + (size_t)j * CLASS_NUM,
                beta + (size_t)(j + 1) * CLASS_NUM,
                beta + (size_t)j * CLASS_NUM);
        }
    }
}